// NodeModel_64252710748376
// MI455X (gfx1250) — compile-verified
//
#include <hip/hip_runtime.h>

// CDNA5 / gfx1250: wave32, WMMA 16x16xK. We use V_WMMA_F32_16X16X4_F32 to keep
// full fp32 precision (reference is fp32; bf16/f16 would miss tolerance on
// 64/256-deep reductions).

typedef float v2f __attribute__((ext_vector_type(2)));
typedef float v8f __attribute__((ext_vector_type(8)));

#define D_IN   64
#define D_HID  256
#define D_OUT  64
#define LDS_STRIDE (D_HID + 4)   // pad rows -> layer-2 ds_load_b64 hits distinct banks

// ---------------------------------------------------------------------------
// Pass 0: zero the per-node denominator accumulator (workspace is not
// re-initialized between replays, so we must do this every call).
__global__ void zero_denom_kernel(float* __restrict__ denom, int n) {
    int i = blockIdx.x * blockDim.x + threadIdx.x;
    if (i < n) denom[i] = 0.0f;
}

// Pass 1: denom[n] = sum of edge_attr over edges with col==n.
// (Reference numerator is edge_attr * x[col] scattered to col -> equals
//  x[n,:] * denom[n]; only denom is needed, plus its ==0 mask for the NaN path.)
__global__ void edge_denom_kernel(const int* __restrict__ col,
                                  const float* __restrict__ edge_attr,
                                  float* __restrict__ denom, int E) {
    int e = blockIdx.x * blockDim.x + threadIdx.x;
    if (e < E) atomicAdd(&denom[col[e]], edge_attr[e]);
}

// ---------------------------------------------------------------------------
// Pass 2: fused geo_agg + MLP. One block = 8 waves = 32 node rows.
//   geo_agg[n,:] = x[n,:] * (denom[n] != 0 ? 2 : 1)   (denom/denom == 1.0 exactly)
//   h  = relu(geo_agg @ W1 + b1)   -> LDS (32 x 256, padded)
//   out = relu(h @ W2 + b2)
//
// WMMA fp32 register layouts (ISA 7.12.2):
//   A 16x4 : lane L<16 -> row L, VGPR0/1 = K k,k+1 ; lane L+16 -> K k+2,k+3
//   B 4x16 : lane L<16 -> col L, K k,k+1            ; lane L+16 -> col L-16, K k+2,k+3
//   C/D    : VGPR i -> row i (lanes 0-15) / row i+8 (lanes 16-31), col = lane&15
__global__ __launch_bounds__(256) void node_mlp_kernel(
    const float* __restrict__ x, const float* __restrict__ denom,
    const float* __restrict__ W1, const float* __restrict__ b1,
    const float* __restrict__ W2, const float* __restrict__ b2,
    float* __restrict__ out, int N)
{
    __shared__ float hbuf[32 * LDS_STRIDE];

    const int lane  = threadIdx.x & 31;
    const int wave  = threadIdx.x >> 5;
    const int rt    = wave >> 2;            // row tile within block: 0 or 1
    const int cg    = wave & 3;             // column group
    const int mrow  = lane & 15;            // tile row (A) / tile col (B,D)
    const int khalf = (lane >> 4) << 1;     // 0 for lanes 0-15, 2 for lanes 16-31

    const int blockRow0 = blockIdx.x * 32;

    // ---- geo_agg rows as A fragments (16 K-blocks of 4) -------------------
    int grow  = blockRow0 + rt * 16 + mrow;
    int growc = (grow < N) ? grow : 0;              // clamp; fake rows masked at store
    float s     = denom[growc];
    float scale = (s != 0.0f) ? 2.0f : 1.0f;        // NaN path -> keep x

    v2f afrag[16];
    const float* xrow = x + (long)growc * D_IN;
    #pragma unroll
    for (int kb = 0; kb < 16; ++kb) {
        v2f a = *(const v2f*)(xrow + kb * 4 + khalf);
        afrag[kb] = a * scale;
    }

    // ---- layer 1: this wave computes hidden cols [cg*64, cg*64+64) --------
    #pragma unroll
    for (int ct = 0; ct < 4; ++ct) {
        const int c0 = cg * 64 + ct * 16;
        v8f acc = {};
        #pragma unroll
        for (int kb = 0; kb < 16; ++kb) {
            const int krow = kb * 4 + khalf;
            v2f b;
            b.x = W1[(krow    ) * D_HID + c0 + mrow];
            b.y = W1[(krow + 1) * D_HID + c0 + mrow];
            acc = __builtin_amdgcn_wmma_f32_16x16x4_f32(
                false, afrag[kb], false, b, (short)0, acc, false, false);
        }
        const float bias = b1[c0 + mrow];
        #pragma unroll
        for (int i = 0; i < 8; ++i) {
            const int r = rt * 16 + ((lane < 16) ? i : (i + 8));
            const float v = acc[i] + bias;
            hbuf[r * LDS_STRIDE + c0 + mrow] = (v > 0.0f) ? v : 0.0f;
        }
    }

    __syncthreads();

    // ---- layer 2: this wave computes out tile [rt*16 rows, cg*16 cols) ----
    {
        const int c0 = cg * 16;
        v8f acc = {};
        const float* hrow = &hbuf[(rt * 16 + mrow) * LDS_STRIDE];
        #pragma unroll
        for (int kb = 0; kb < 64; ++kb) {
            const int krow = kb * 4 + khalf;
            v2f a = *(const v2f*)(hrow + krow);     // ds_load_b64, 8B aligned
            v2f b;
            b.x = W2[(krow    ) * D_OUT + c0 + mrow];
            b.y = W2[(krow + 1) * D_OUT + c0 + mrow];
            acc = __builtin_amdgcn_wmma_f32_16x16x4_f32(
                false, a, false, b, (short)0, acc, false, false);
        }
        const float bias = b2[c0 + mrow];
        #pragma unroll
        for (int i = 0; i < 8; ++i) {
            const int r = blockRow0 + rt * 16 + ((lane < 16) ? i : (i + 8));
            if (r < N) {
                const float v = acc[i] + bias;
                out[(long)r * D_OUT + c0 + mrow] = (v > 0.0f) ? v : 0.0f;
            }
        }
    }
}

// ---------------------------------------------------------------------------
extern "C" void kernel_launch(void* const* d_in, const int* in_sizes, int n_in,
                              void* d_out, int out_size, void* d_ws, size_t ws_size,
                              hipStream_t stream) {
    const float* x          = (const float*)d_in[0];
    const int*   edge_index = (const int*)  d_in[1];   // [2, E]: row then col
    const float* edge_attr  = (const float*)d_in[2];   // [E, 1]
    // d_in[3] = u, d_in[4] = batch: unused by the reference computation
    const float* W1 = (const float*)d_in[5];
    const float* b1 = (const float*)d_in[6];
    const float* W2 = (const float*)d_in[7];
    const float* b2 = (const float*)d_in[8];

    const int N = in_sizes[0] / D_IN;
    const int E = in_sizes[2];
    const int* col = edge_index + E;                   // second row of edge_index

    float* denom = (float*)d_ws;                       // N floats of scratch

    zero_denom_kernel<<<(N + 255) / 256, 256, 0, stream>>>(denom, N);
    edge_denom_kernel<<<(E + 255) / 256, 256, 0, stream>>>(col, edge_attr, denom, E);
    node_mlp_kernel<<<(N + 31) / 32, 256, 0, stream>>>(
        x, denom, W1, b1, W2, b2, (float*)d_out, N);
}